// SkeletalConv_49658411877082
// MI455X (gfx1250) — compile-verified
//
#include <hip/hip_runtime.h>

// Skeletal graph Conv1d -> per-edge mean, mapped onto V_WMMA_F32_16X16X4_F32.
// x: (CIN=16, J=25, L=65536) f32
// W: (P=48, COUT=16, CIN=16, K=3) f32
// b: (P=48, COUT=16) f32
// out: (COUT=16, E=24, LOUT=65534) f32
//
// Each wave computes a 16(co) x 16(t) tile for one edge e, accumulating both
// neighbor pairs (p0=2e uses joint e, p1=2e+1 uses joint e+1) with 24
// v_wmma_f32_16x16x4_f32 per tile (reduction dim R = CIN*K = 48 = 12 chunks
// of 4, times 2 pairs).

typedef float v2f __attribute__((ext_vector_type(2)));
typedef float v8f __attribute__((ext_vector_type(8)));

#define CIN    16
#define JNT    25
#define EDG    24
#define COUTC  16
#define KW     3
#define LIN    65536
#define LOUTC  (LIN - KW + 1)      // 65534
#define RDIM   (CIN * KW)          // 48
#define NCHUNK (RDIM / 4)          // 12
#define NTILE  ((LOUTC + 15) / 16) // 4096

__global__ __launch_bounds__(128) void skel_conv_wmma_f32(
    const float* __restrict__ x,
    const float* __restrict__ W,
    const float* __restrict__ b,
    float* __restrict__ out)
{
    const int lane = threadIdx.x & 31;
    const int wave = threadIdx.x >> 5;
    const int e    = blockIdx.y;          // edge 0..23
    const int h    = lane >> 4;           // lane half: 0 or 1
    const int mn   = lane & 15;           // A-row (co) / B-col (t) index

    const int p0 = 2 * e;
    const int p1 = 2 * e + 1;

    // ---- A fragments: weights for both pairs, reduction index r = k*16 + ci.
    // A(16x4 f32) layout: vgpr0 = K=4c+2h, vgpr1 = K=4c+2h+1 for lane 16h+m.
    v2f A0[NCHUNK], A1[NCHUNK];
    #pragma unroll
    for (int c = 0; c < NCHUNK; ++c) {
        const int r0 = 4 * c + 2 * h;
        const int r1 = r0 + 1;
        const int ci0 = r0 & 15, k0 = r0 >> 4;
        const int ci1 = r1 & 15, k1 = r1 >> 4;
        A0[c].x = W[((p0 * COUTC + mn) * CIN + ci0) * KW + k0];
        A0[c].y = W[((p0 * COUTC + mn) * CIN + ci1) * KW + k1];
        A1[c].x = W[((p1 * COUTC + mn) * CIN + ci0) * KW + k0];
        A1[c].y = W[((p1 * COUTC + mn) * CIN + ci1) * KW + k1];
    }

    // ---- bias term per output element: co = 8h + r
    float bias[8];
    #pragma unroll
    for (int r = 0; r < 8; ++r) {
        const int co = 8 * h + r;
        bias[r] = 0.5f * (b[p0 * COUTC + co] + b[p1 * COUTC + co]);
    }

    const int gw      = blockIdx.x * 4 + wave;  // global wave id along time
    const int gstride = gridDim.x * 4;

    for (int tile = gw; tile < NTILE; tile += gstride) {
        const int t0 = tile * 16;
        v8f acc0 = {};
        v8f acc1 = {};

        if (t0 + 16 + (KW - 1) <= LIN) {
            // -------- fast path: full tile, unguarded loads & stores --------
            #pragma unroll
            for (int c = 0; c < NCHUNK; ++c) {
                const int r0 = 4 * c + 2 * h;
                const int r1 = r0 + 1;
                const int ta = t0 + mn + (r0 >> 4);   // t + k for row r0
                const int tb = t0 + mn + (r1 >> 4);   // t + k for row r1
                // B(4x16 f32) layout: vgpr0 = K-row 4c+2h, vgpr1 = 4c+2h+1
                v2f B0, B1;
                B0.x = x[((r0 & 15) * JNT + e) * LIN + ta];
                B0.y = x[((r1 & 15) * JNT + e) * LIN + tb];
                B1.x = x[((r0 & 15) * JNT + (e + 1)) * LIN + ta];
                B1.y = x[((r1 & 15) * JNT + (e + 1)) * LIN + tb];
                acc0 = __builtin_amdgcn_wmma_f32_16x16x4_f32(
                    false, A0[c], false, B0, (short)0, acc0, false, false);
                acc1 = __builtin_amdgcn_wmma_f32_16x16x4_f32(
                    false, A1[c], false, B1, (short)0, acc1, false, false);
            }
            #pragma unroll
            for (int r = 0; r < 8; ++r) {
                const int co = 8 * h + r;
                out[(co * EDG + e) * LOUTC + t0 + mn] =
                    0.5f * (acc0[r] + acc1[r]) + bias[r];
            }
        } else {
            // -------- tail tile: clamp loads, guard stores --------
            #pragma unroll
            for (int c = 0; c < NCHUNK; ++c) {
                const int r0 = 4 * c + 2 * h;
                const int r1 = r0 + 1;
                int ta = t0 + mn + (r0 >> 4);
                int tb = t0 + mn + (r1 >> 4);
                ta = (ta < LIN) ? ta : (LIN - 1);
                tb = (tb < LIN) ? tb : (LIN - 1);
                v2f B0, B1;
                B0.x = x[((r0 & 15) * JNT + e) * LIN + ta];
                B0.y = x[((r1 & 15) * JNT + e) * LIN + tb];
                B1.x = x[((r0 & 15) * JNT + (e + 1)) * LIN + ta];
                B1.y = x[((r1 & 15) * JNT + (e + 1)) * LIN + tb];
                acc0 = __builtin_amdgcn_wmma_f32_16x16x4_f32(
                    false, A0[c], false, B0, (short)0, acc0, false, false);
                acc1 = __builtin_amdgcn_wmma_f32_16x16x4_f32(
                    false, A1[c], false, B1, (short)0, acc1, false, false);
            }
            if (t0 + mn < LOUTC) {
                #pragma unroll
                for (int r = 0; r < 8; ++r) {
                    const int co = 8 * h + r;
                    out[(co * EDG + e) * LOUTC + t0 + mn] =
                        0.5f * (acc0[r] + acc1[r]) + bias[r];
                }
            }
        }
    }
}

extern "C" void kernel_launch(void* const* d_in, const int* in_sizes, int n_in,
                              void* d_out, int out_size, void* d_ws, size_t ws_size,
                              hipStream_t stream) {
    (void)in_sizes; (void)n_in; (void)out_size; (void)d_ws; (void)ws_size;
    const float* x = (const float*)d_in[0];
    const float* W = (const float*)d_in[1];
    const float* b = (const float*)d_in[2];
    float* out = (float*)d_out;

    dim3 grid(32, EDG);   // 32 tile-groups x 24 edges
    dim3 block(128);      // 4 waves (wave32)
    hipLaunchKernelGGL(skel_conv_wmma_f32, grid, block, 0, stream, x, W, b, out);
}